// TopKSpatialPool2d_45878840656452
// MI455X (gfx1250) — compile-verified
//
#include <hip/hip_runtime.h>
#include <cfloat>
#include <cstdint>

// Problem constants (from reference: x = [32, 512, 56, 56] fp32, K = 10)
#define HW_     3136        // 56*56
#define C_      512
#define PIX_    224         // pixels per block (HW_ = 14 * 224, never crosses batch)
#define CH_     32          // channels per LDS chunk
#define NCHUNK_ (C_ / CH_)  // 16
#define SWEEPS_ 8           // async b128 issues per thread per chunk
#define KTOP_   10
#define NBUF_   3           // triple buffer: 2 chunks always in flight

__device__ __forceinline__ void issue_chunk_async(const float* __restrict__ gimg,
                                                  float (*lbuf)[CH_ * PIX_],
                                                  int k, int c_in_sweep, int pix4) {
#pragma unroll
    for (int sw = 0; sw < SWEEPS_; ++sw) {
        const int cl = sw * 4 + c_in_sweep;                       // 0..31
        const float* g = gimg + (size_t)k * CH_ * HW_ + (size_t)cl * HW_ + pix4;
        const unsigned loff =
            (unsigned)(unsigned long long)&lbuf[k % NBUF_][cl * PIX_ + pix4];
        asm volatile("global_load_async_to_lds_b128 %0, %1, off"
                     :: "v"(loff), "v"(g) : "memory");
    }
}

__global__ __launch_bounds__(PIX_) void topk_pool_kernel(const float* __restrict__ in,
                                                         float* __restrict__ out) {
    // Triple-buffered staging: [3][CH_ * PIX_] floats = 86,016 B LDS
    __shared__ float lbuf[NBUF_][CH_ * PIX_];

    const int tid     = threadIdx.x;
    const int img     = blockIdx.y;              // batch index
    const int pixbase = blockIdx.x * PIX_;       // spatial base within image
    const float* gimg = in + (size_t)img * C_ * HW_ + pixbase;

    // Copy-phase mapping: thread -> (channel-in-sweep, 4-pixel group)
    const int c_in_sweep = tid / 56;             // 0..3
    const int pix4       = (tid % 56) * 4;       // 0..220 (16B aligned)

    // ---- Prologue: put two chunks in flight ----
    issue_chunk_async(gimg, lbuf, 0, c_in_sweep, pix4);
    issue_chunk_async(gimg, lbuf, 1, c_in_sweep, pix4);

    // Running top-10, sorted descending: m[0] >= ... >= m[9]
    float m[KTOP_];
#pragma unroll
    for (int i = 0; i < KTOP_; ++i) m[i] = -FLT_MAX;

#pragma unroll 1
    for (int k = 0; k < NCHUNK_; ++k) {
        if (k + 2 < NCHUNK_) {
            // Keep two chunks in flight; buf[(k+2)%3] was drained before
            // iteration (k-1)'s trailing barrier.
            issue_chunk_async(gimg, lbuf, k + 2, c_in_sweep, pix4);
            asm volatile("s_wait_asynccnt 16" ::: "memory");  // chunk k resident
        } else if (k + 1 < NCHUNK_) {
            asm volatile("s_wait_asynccnt 8" ::: "memory");   // only k+1 pending
        } else {
            asm volatile("s_wait_asynccnt 0" ::: "memory");
        }
        __syncthreads();   // all waves' chunk-k data visible

        const float* buf = lbuf[k % NBUF_];
#pragma unroll 8
        for (int cl = 0; cl < CH_; ++cl) {
            const float v = buf[cl * PIX_ + tid];
            // Wave32 vote: skip insert machinery if no lane beats current 10th
            if (__builtin_amdgcn_ballot_w32(v > m[KTOP_ - 1]) != 0u) {
                float key = (v > m[KTOP_ - 1]) ? v : -FLT_MAX;  // no-op sentinel
#pragma unroll
                for (int i = 0; i < KTOP_; ++i) {               // branchless bubble
                    const float hi = fmaxf(m[i], key);
                    key  = fminf(m[i], key);
                    m[i] = hi;
                }
            }
        }
        __syncthreads();   // buffer drained before it is overwritten (iter k+1 issue)
    }

    float s = 0.0f;
#pragma unroll
    for (int i = 0; i < KTOP_; ++i) s += m[i];
    out[(size_t)img * HW_ + pixbase + tid] = s * 0.1f;
}

extern "C" void kernel_launch(void* const* d_in, const int* in_sizes, int n_in,
                              void* d_out, int out_size, void* d_ws, size_t ws_size,
                              hipStream_t stream) {
    const float* in = (const float*)d_in[0];
    float* out = (float*)d_out;
    const int B = in_sizes[0] / (C_ * HW_);      // 32
    dim3 grid(HW_ / PIX_, B);                    // 14 x 32 = 448 blocks
    topk_pool_kernel<<<grid, PIX_, 0, stream>>>(in, out);
}